// Conv2d_68298569941797
// MI455X (gfx1250) — compile-verified
//
#include <hip/hip_runtime.h>
#include <hip/hip_bf16.h>

typedef float v2f __attribute__((ext_vector_type(2)));
typedef float v8f __attribute__((ext_vector_type(8)));

#define IN_H   224
#define IN_W   224
#define OUT_H  222
#define OUT_W  222
#define BATCH  32
#define OCH    64
#define XTILES 14                       // ceil(222/16)
#define NTILES (BATCH * OUT_H * XTILES) // one wave per (b, oy, 16-wide x tile)

// im2col element: data[b, oy+ky, x+kx] for flattened k = ky*3+kx, zero-padded for k>=9.
__device__ __forceinline__ float im2col_elem(const float* __restrict__ in,
                                             int b, int oy, int x, int k) {
    if (k >= 9) return 0.0f;
    int ky = k / 3;
    int kx = k - ky * 3;
    int xi = x + kx;
    if (xi > IN_W - 1) xi = IN_W - 1;   // clamp (lanes at x>=222 never store)
    return in[((size_t)b * IN_H + (oy + ky)) * IN_W + xi];
}

__device__ __forceinline__ float w_elem(const float* __restrict__ w, int ch, int k) {
    return (k < 9) ? w[ch * 9 + k] : 0.0f;
}

__global__ __launch_bounds__(256)
void conv3x3_wmma_f32(const float* __restrict__ in,
                      const float* __restrict__ wgt,
                      float* __restrict__ out) {
    const int lane  = threadIdx.x & 31;
    const int wave  = blockIdx.x * (blockDim.x >> 5) + (threadIdx.x >> 5);
    if (wave >= NTILES) return;          // wave-uniform: EXEC stays all-ones

    const int xt  = wave % XTILES;
    const int t2  = wave / XTILES;
    const int oy  = t2 % OUT_H;
    const int b   = t2 / OUT_H;
    const int ox0 = xt * 16;

    const int lx    = lane & 15;         // N (x) / M (ch) index within 16
    const int lhalf = lane >> 4;         // selects K pair {0,1} vs {2,3}
    const int x     = ox0 + lx;

    // ---- B matrix (im2col, 4x16 per K-chunk): N=x across lanes, K across (vgpr, half)
    v2f Bm[3];
#pragma unroll
    for (int c = 0; c < 3; ++c) {
        const int k0 = c * 4 + lhalf * 2;
        Bm[c][0] = im2col_elem(in, b, oy, x, k0);
        Bm[c][1] = im2col_elem(in, b, oy, x, k0 + 1);
    }

    // ---- A matrices (weights, 16x4 per chunk), 4 channel tiles of 16
    v2f Am[4][3];
#pragma unroll
    for (int j = 0; j < 4; ++j) {
        const int ch = j * 16 + lx;
#pragma unroll
        for (int c = 0; c < 3; ++c) {
            const int k0 = c * 4 + lhalf * 2;
            Am[j][c][0] = w_elem(wgt, ch, k0);
            Am[j][c][1] = w_elem(wgt, ch, k0 + 1);
        }
    }

    // ---- D = A x B (+C): 4 tiles x 3 K-steps = 12 WMMAs
    v8f acc[4] = {v8f{}, v8f{}, v8f{}, v8f{}};
#pragma unroll
    for (int j = 0; j < 4; ++j) {
#pragma unroll
        for (int c = 0; c < 3; ++c) {
            acc[j] = __builtin_amdgcn_wmma_f32_16x16x4_f32(
                /*neg_a=*/false, Am[j][c],
                /*neg_b=*/false, Bm[c],
                /*c_mod=*/(short)0, acc[j],
                /*reuse_a=*/false, /*reuse_b=*/false);
        }
    }

    // ---- Store: D layout -> lane half gives ch+8, vgpr r gives ch, lanes give x.
    // 16 consecutive lanes write 16 consecutive floats (64B runs) -> coalesced NT stream.
    const bool xvalid = (x < OUT_W);
#pragma unroll
    for (int j = 0; j < 4; ++j) {
#pragma unroll
        for (int r = 0; r < 8; ++r) {
            const int ch = j * 16 + lhalf * 8 + r;
            if (xvalid) {
                const size_t idx = (((size_t)b * OCH + ch) * OUT_H + oy) * OUT_W + x;
                __builtin_nontemporal_store(acc[j][r], out + idx);
            }
        }
    }
}

extern "C" void kernel_launch(void* const* d_in, const int* in_sizes, int n_in,
                              void* d_out, int out_size, void* d_ws, size_t ws_size,
                              hipStream_t stream) {
    const float* data = (const float*)d_in[0];   // [32,1,224,224]
    const float* wgt  = (const float*)d_in[1];   // [64,1,3,3]
    float* out        = (float*)d_out;           // [32,64,222,222]

    const int waves_per_block = 256 / 32;        // 8 waves per 256-thread block
    const int blocks = (NTILES + waves_per_block - 1) / waves_per_block;
    conv3x3_wmma_f32<<<blocks, 256, 0, stream>>>(data, wgt, out);
}